// NeuralODE_28415503631103
// MI455X (gfx1250) — compile-verified
//
#include <hip/hip_runtime.h>
#include <math.h>

typedef __attribute__((ext_vector_type(2))) float v2f;
typedef __attribute__((ext_vector_type(4))) float v4f;
typedef __attribute__((ext_vector_type(8))) float v8f;

#define B_ 256
#define F_ 256
#define HD_ 1024
#define NX (B_ * F_)    /* 65536 */
#define NH (B_ * HD_)   /* 262144 */
#define N_ITER 64
#define KC 64           /* K-chunk staged in LDS */
#define PSTRIDE 160     /* floats per K-pair row in LDS; 160 mod 64 = 32 ->  */
                        /* the two half-wave K-groups land on disjoint banks */

// ---- Dormand-Prince 5(4) tableau (fp32, computed in double then truncated,
// matching JAX weak-type f32 promotion of Python float scalars) ----
__constant__ float cA[7][7] = {
  {0,0,0,0,0,0,0},
  {(float)(1.0/5.0),0,0,0,0,0,0},
  {(float)(3.0/40.0),(float)(9.0/40.0),0,0,0,0,0},
  {(float)(44.0/45.0),(float)(-56.0/15.0),(float)(32.0/9.0),0,0,0,0},
  {(float)(19372.0/6561.0),(float)(-25360.0/2187.0),(float)(64448.0/6561.0),(float)(-212.0/729.0),0,0,0},
  {(float)(9017.0/3168.0),(float)(-355.0/33.0),(float)(46732.0/5247.0),(float)(49.0/176.0),(float)(-5103.0/18656.0),0,0},
  {(float)(35.0/384.0),0.0f,(float)(500.0/1113.0),(float)(125.0/192.0),(float)(-2187.0/6784.0),(float)(11.0/84.0),0}
};
__constant__ float cC[7] = {0.0f,(float)(1.0/5.0),(float)(3.0/10.0),(float)(4.0/5.0),(float)(8.0/9.0),1.0f,1.0f};
__constant__ float cB5[7] = {(float)(35.0/384.0),0.0f,(float)(500.0/1113.0),(float)(125.0/192.0),(float)(-2187.0/6784.0),(float)(11.0/84.0),0.0f};
__constant__ float cD[7] = {
  (float)(35.0/384.0 - 5179.0/57600.0),
  0.0f,
  (float)(500.0/1113.0 - 7571.0/16695.0),
  (float)(125.0/192.0 - 393.0/640.0),
  (float)(-2187.0/6784.0 + 92097.0/339200.0),
  (float)(11.0/84.0 - 187.0/2100.0),
  (float)(-1.0/40.0)
};

// ---------------- init: X <- x0, out[0:NX] <- x0, scalars reset ----------------
__global__ void k_init(const float* __restrict__ x0, float* __restrict__ X,
                       float* __restrict__ S, float* __restrict__ out) {
  int e = blockIdx.x * blockDim.x + threadIdx.x;
  if (e < NX) { float v = x0[e]; X[e] = v; out[e] = v; }
  if (blockIdx.x == 0 && threadIdx.x == 0) { S[0] = 0.0f; S[1] = 0.05f; S[2] = 0.0f; }
}

// ---------------- begin iteration: dt_c = clamp(min(dt, 1-t), 0, inf) ----------
__global__ void k_begin(float* __restrict__ S) {
  if (threadIdx.x == 0) {
    float t = S[0], dt = S[1];
    S[2] = fmaxf(fminf(dt, 1.0f - t), 0.0f);
  }
}

// ---------------- stage input: XI = X + dt_c * sum_j A[i][j] * KS[j] -----------
__global__ void k_combine(const float* __restrict__ X, const float* __restrict__ KS,
                          const float* __restrict__ S, float* __restrict__ XI, int stage) {
  int e = blockIdx.x * blockDim.x + threadIdx.x;
  float dtc = S[2];
  float v = X[e];
  for (int j = 0; j < stage; ++j) {
    float a = cA[stage][j];
    if (a != 0.0f) v += (dtc * a) * KS[j * NX + e];
  }
  XI[e] = v;
}

// Shared GEMM core: block = 8 waves = 128 output rows x 64 output cols.
// B panel (KC x 64) staged once per chunk into LDS in K-pair-interleaved form:
//   Bs[(k>>1)*PSTRIDE + n*2 + (k&1)]
// so a B fragment {B[k][n], B[k+1][n]} is one aligned ds_load_b64 straight into
// an even VGPR pair (no register shuffles before v_wmma).
// Per-lane fragment ownership follows the CDNA5 f32 WMMA layouts:
//   A 16x4:  lane(r,h) VGPR v = A[m0+r][k + 2h + v]
//   B 4x16:  lane(r,h) VGPR v = B[k + 2h + v][n]
//   C 16x16: lane(r,h) VGPR v = C[m0 + v + 8h][n]
#define GEMM_CORE(AROW_BASE, BGLOBAL, LDGB, KTOT)                              \
  __shared__ float Bs[(KC / 2) * PSTRIDE];                                     \
  const int tid  = threadIdx.x;                                                \
  const int wv   = tid >> 5;                                                   \
  const int lane = tid & 31;                                                   \
  const int row  = lane & 15;                                                  \
  const int kh   = lane >> 4;                                                  \
  const int m0 = blockIdx.y * 128 + wv * 16;                                   \
  const int n0 = blockIdx.x * 64;                                              \
  const int tcol = (tid & 15) * 4;                                             \
  const int trow = tid >> 4;                                                   \
  v8f acc0 = {}, acc1 = {}, acc2 = {}, acc3 = {};                              \
  for (int k0 = 0; k0 < (KTOT); k0 += KC) {                                    \
    for (int p = 0; p < KC; p += 16) {                                         \
      const int kr = trow + p;                                                 \
      const float* gp = (BGLOBAL) + (size_t)(k0 + kr) * (LDGB) + n0 + tcol;    \
      v4f bv = *(const v4f*)gp;                                                \
      float* lp = Bs + (kr >> 1) * PSTRIDE + tcol * 2 + (kr & 1);              \
      lp[0] = bv.x; lp[2] = bv.y; lp[4] = bv.z; lp[6] = bv.w;                  \
    }                                                                          \
    __syncthreads();                                                           \
    const float* arow = (AROW_BASE) + k0;                                      \
    for (int kk = 0; kk < KC; kk += 4) {                                       \
      v2f a;                                                                   \
      a.x = arow[kk + 2 * kh];                                                 \
      a.y = arow[kk + 2 * kh + 1];                                             \
      const float* bp = Bs + ((kk >> 1) + kh) * PSTRIDE + row * 2;             \
      v2f b0 = *(const v2f*)(bp + 0);                                          \
      v2f b1v = *(const v2f*)(bp + 32);                                        \
      v2f b2v = *(const v2f*)(bp + 64);                                        \
      v2f b3v = *(const v2f*)(bp + 96);                                        \
      acc0 = __builtin_amdgcn_wmma_f32_16x16x4_f32(false, a, false, b0,  (short)0, acc0, false, false); \
      acc1 = __builtin_amdgcn_wmma_f32_16x16x4_f32(false, a, false, b1v, (short)0, acc1, false, false); \
      acc2 = __builtin_amdgcn_wmma_f32_16x16x4_f32(false, a, false, b2v, (short)0, acc2, false, false); \
      acc3 = __builtin_amdgcn_wmma_f32_16x16x4_f32(false, a, false, b3v, (short)0, acc3, false, false); \
    }                                                                          \
    __syncthreads();                                                           \
  }

// ---------------- layer 1: Hb = tanh(XI @ W1[:-1] + t_i*W1[-1] + b1) -----------
// grid (1024/64, 256/128) = (16, 2), block 256.
__global__ void k_layer1(const float* __restrict__ XI, const float* __restrict__ W1,
                         const float* __restrict__ b1, const float* __restrict__ S,
                         float* __restrict__ Hb, int stage) {
  const float t_i = S[0] + cC[stage] * S[2];
  GEMM_CORE(XI + (size_t)(m0 + row) * F_, W1, HD_, F_)
  const float* wt = W1 + (size_t)F_ * HD_;  // last row of W1 (time weights)
#pragma unroll
  for (int v = 0; v < 8; ++v) {
    const int m = m0 + v + 8 * kh;
    int n;
    n = n0 + row;      Hb[(size_t)m * HD_ + n] = tanhf(acc0[v] + t_i * wt[n] + b1[n]);
    n = n0 + 16 + row; Hb[(size_t)m * HD_ + n] = tanhf(acc1[v] + t_i * wt[n] + b1[n]);
    n = n0 + 32 + row; Hb[(size_t)m * HD_ + n] = tanhf(acc2[v] + t_i * wt[n] + b1[n]);
    n = n0 + 48 + row; Hb[(size_t)m * HD_ + n] = tanhf(acc3[v] + t_i * wt[n] + b1[n]);
  }
}

// ---------------- layer 2: K = Hb @ W2 + b2. grid (256/64, 256/128) = (4, 2) ---
__global__ void k_layer2(const float* __restrict__ Hb, const float* __restrict__ W2,
                         const float* __restrict__ b2, float* __restrict__ K) {
  GEMM_CORE(Hb + (size_t)(m0 + row) * HD_, W2, F_, HD_)
#pragma unroll
  for (int v = 0; v < 8; ++v) {
    const int m = m0 + v + 8 * kh;
    int n;
    n = n0 + row;      K[(size_t)m * F_ + n] = acc0[v] + b2[n];
    n = n0 + 16 + row; K[(size_t)m * F_ + n] = acc1[v] + b2[n];
    n = n0 + 32 + row; K[(size_t)m * F_ + n] = acc2[v] + b2[n];
    n = n0 + 48 + row; K[(size_t)m * F_ + n] = acc3[v] + b2[n];
  }
}

// ---------------- epilogue: x5/err, err_norm reduction, accept, state update ---
__global__ void k_epilogue(float* __restrict__ X, const float* __restrict__ KS,
                           float* __restrict__ X5, float* __restrict__ S) {
  __shared__ float red[1024];
  __shared__ float bc;
  const int tid = threadIdx.x;
  const float t = S[0], dt = S[1], dtc = S[2];
  const bool done = (t >= 1.0f);

  float partial = 0.0f;
  for (int e = tid; e < NX; e += 1024) {
    float xv = X[e];
    float x5 = xv, err = 0.0f;
#pragma unroll
    for (int j = 0; j < 7; ++j) {
      float kj = KS[j * NX + e];
      float b5 = cB5[j];
      if (b5 != 0.0f) x5 += (dtc * b5) * kj;
      float d = cD[j];
      if (d != 0.0f) err += (dtc * d) * kj;
    }
    X5[e] = x5;
    float scale = 1e-4f + 1e-3f * fmaxf(fabsf(xv), fabsf(x5));
    float r = err / scale;
    partial += r * r;
  }
  red[tid] = partial;
  __syncthreads();
  for (int s = 512; s > 0; s >>= 1) {
    if (tid < s) red[tid] += red[tid + s];
    __syncthreads();
  }
  if (tid == 0) {
    float errn = sqrtf(red[0] / (float)NX);
    bool accept = (errn <= 1.0f);
    float factor = 0.9f * powf(errn + 1e-10f, -0.2f);
    factor = fminf(fmaxf(factor, 0.2f), 5.0f);
    S[0] = done ? t : (accept ? t + dtc : t);
    S[1] = done ? dt : dtc * factor;
    bc = (accept && !done) ? 1.0f : 0.0f;
  }
  __syncthreads();
  if (bc != 0.0f) {
    for (int e = tid; e < NX; e += 1024) X[e] = X5[e];
  }
}

// ---------------- finalize: out[NX:2NX] <- X --------------------------------
__global__ void k_final(const float* __restrict__ X, float* __restrict__ out) {
  int e = blockIdx.x * blockDim.x + threadIdx.x;
  out[e] = X[e];
}

extern "C" void kernel_launch(void* const* d_in, const int* in_sizes, int n_in,
                              void* d_out, int out_size, void* d_ws, size_t ws_size,
                              hipStream_t stream) {
  const float* x0 = (const float*)d_in[0];
  const float* W1 = (const float*)d_in[1];
  const float* b1 = (const float*)d_in[2];
  const float* W2 = (const float*)d_in[3];
  const float* b2 = (const float*)d_in[4];
  float* out = (float*)d_out;
  float* ws  = (float*)d_ws;

  float* X  = ws;                        // 65536
  float* XI = ws + NX;                   // 65536
  float* X5 = ws + 2 * NX;               // 65536
  float* Hb = ws + 3 * NX;               // 262144
  float* KS = ws + 3 * NX + NH;          // 7 * 65536
  float* S  = ws + 3 * NX + NH + 7 * NX; // scalars: t, dt, dt_c

  k_init<<<NX / 256, 256, 0, stream>>>(x0, X, S, out);
  for (int it = 0; it < N_ITER; ++it) {
    k_begin<<<1, 32, 0, stream>>>(S);
    for (int st = 0; st < 7; ++st) {
      k_combine<<<NX / 256, 256, 0, stream>>>(X, KS, S, XI, st);
      k_layer1<<<dim3(HD_ / 64, B_ / 128), 256, 0, stream>>>(XI, W1, b1, S, Hb, st);
      k_layer2<<<dim3(F_ / 64, B_ / 128), 256, 0, stream>>>(Hb, W2, b2, KS + st * NX);
    }
    k_epilogue<<<1, 1024, 0, stream>>>(X, KS, X5, S);
  }
  k_final<<<NX / 256, 256, 0, stream>>>(X, out + NX);
}